// GIN_dc_63771674411496
// MI455X (gfx1250) — compile-verified
//
#include <hip/hip_runtime.h>
#include <math.h>
#include <stdint.h>

// ---------------------------------------------------------------------------
// GIN+GCN GNN forward for MI455X (gfx1250), fp32 end-to-end.
//  * Dense node GEMMs: V_WMMA_F32_16X16X4_F32, 80x128 stripe per block,
//    B (and A) staged in LDS by the Tensor Data Mover (tensor_load_to_lds),
//    using TDM padding to produce a bank-conflict-free 132-float row stride.
//  * Edge aggregation: coalesced gathers + global float atomics (HBM-bound).
// ---------------------------------------------------------------------------

typedef __attribute__((ext_vector_type(2))) float v2f;
typedef __attribute__((ext_vector_type(8))) float v8f;
typedef __attribute__((ext_vector_type(4))) unsigned int v4u;
typedef __attribute__((ext_vector_type(8))) unsigned int v8u;

#define HH   128   // hidden size H
#define PADW 132   // LDS row stride in floats (128 + 4 pad dwords from TDM)
#define MSUB 5     // 16-row sub-tiles per wave (80 rows per block)

__device__ __forceinline__ v8f wmma4(v2f a, v2f b, v8f c) {
  // D(16x16) = A(16x4) * B(4x16) + C, fp32
  return __builtin_amdgcn_wmma_f32_16x16x4_f32(false, a, false, b, (short)0, c,
                                               false, false);
}

// --- TDM: load a 2D fp32 tile (tile0 x tile1 elements, row stride `stride0`
// elements in memory) into LDS at byte offset lds_off, inserting 4 pad dwords
// after every 128 dwords (-> 132-float LDS row stride). Issued by one lane;
// EXEC is ignored by tensor ops. Tracked by TENSORcnt.
__device__ __forceinline__ void tdm_load_2d(unsigned lds_off,
                                            const float* gptr,
                                            unsigned dim0, unsigned dim1,
                                            unsigned tile0, unsigned tile1,
                                            unsigned stride0) {
  unsigned long long ga = (unsigned long long)(uintptr_t)gptr;
  v4u g0;
  g0[0] = 1u;                                  // count=1, user descriptor
  g0[1] = lds_off;                             // LDS byte address
  g0[2] = (unsigned)(ga & 0xFFFFFFFFu);        // global_addr[31:0]
  g0[3] = (unsigned)((ga >> 32) & 0x1FFFFFFu)  // global_addr[56:32]
          | (2u << 30);                        // type = 2 ("image")
  v8u g1;
  g1[0] = (2u << 16)      // data_size = 4 bytes
        | (1u << 20)      // pad_enable
        | (6u << 22)      // pad_interval: 128 dwords
        | (3u << 25);     // pad_amount: 4 dwords
  g1[1] = (dim0 & 0xFFFFu) << 16;                              // tensor_dim0 lo
  g1[2] = ((dim0 >> 16) & 0xFFFFu) | ((dim1 & 0xFFFFu) << 16); // dim0 hi|dim1 lo
  g1[3] = ((dim1 >> 16) & 0xFFFFu) | ((tile0 & 0xFFFFu) << 16);// dim1 hi|tile0
  g1[4] = (tile1 & 0xFFFFu);                                   // tile1, tile2=0
  g1[5] = stride0;                                             // dim0_stride lo
  g1[6] = 0u;
  g1[7] = 0u;
  asm volatile("tensor_load_to_lds %0, %1" :: "s"(g0), "s"(g1) : "memory");
}

// ---- D[80x128 stripe] = act(A[N,128] @ B[128,128] + bias) ------------------
// grid.x = N/80; block = 256 threads = 8 waves (one 16-col slice per wave).
__global__ __launch_bounds__(256) void gemm128(
    const float* __restrict__ A, const float* __restrict__ B,
    const float* __restrict__ bias, float* __restrict__ D, int act) {
  __shared__ float Bs[HH * PADW];       // 128 x 132 fp32
  __shared__ float As[16 * MSUB * PADW];// 80 x 132 fp32
  const int m0 = blockIdx.x * (16 * MSUB);

  if (threadIdx.x == 0) {
    tdm_load_2d((unsigned)(uintptr_t)&Bs[0], B, HH, HH, HH, HH, HH);
    tdm_load_2d((unsigned)(uintptr_t)&As[0], A + (size_t)m0 * HH, HH,
                16 * MSUB, HH, 16 * MSUB, HH);
    __builtin_amdgcn_s_wait_tensorcnt(0);
  }
  __syncthreads();

  const int lane = threadIdx.x & 31;
  const int lo = lane & 15, hi = lane >> 4;
  const int n0 = (threadIdx.x >> 5) << 4;

  v8f acc[MSUB];
#pragma unroll
  for (int t = 0; t < MSUB; ++t) acc[t] = (v8f){};

#pragma unroll 4
  for (int k = 0; k < HH; k += 4) {
    const int kk = k + 2 * hi;
    v2f b = {Bs[kk * PADW + n0 + lo], Bs[kk * PADW + PADW + n0 + lo]};
#pragma unroll
    for (int t = 0; t < MSUB; ++t) {
      v2f a = *(const v2f*)&As[(16 * t + lo) * PADW + kk];
      acc[t] = wmma4(a, b, acc[t]);
    }
  }

  float bv = bias ? bias[n0 + lo] : 0.0f;
#pragma unroll
  for (int t = 0; t < MSUB; ++t)
#pragma unroll
    for (int j = 0; j < 8; ++j) {
      float v = acc[t][j] + bv;
      if (act) v = fmaxf(v, 0.0f);
      D[(size_t)(m0 + 16 * t + j + 8 * hi) * HH + n0 + lo] = v;
    }
}

// ---- D = act( [X(+AX) | S(+AS)] (N x 256) @ B(256x128) + bias ) ------------
// B staged in LDS via TDM; A composed on the fly from global (L2-resident).
__global__ __launch_bounds__(256) void gemm_cat(
    const float* __restrict__ X, const float* __restrict__ AX,
    const float* __restrict__ S, const float* __restrict__ AS,
    const float* __restrict__ B, const float* __restrict__ bias,
    float* __restrict__ D, int act) {
  __shared__ float Bs[2 * HH * PADW];   // 256 x 132 fp32
  const int m0 = blockIdx.x * (16 * MSUB);

  if (threadIdx.x == 0) {
    tdm_load_2d((unsigned)(uintptr_t)&Bs[0], B, HH, 2 * HH, HH, 2 * HH, HH);
    __builtin_amdgcn_s_wait_tensorcnt(0);
  }
  __syncthreads();

  const int lane = threadIdx.x & 31;
  const int lo = lane & 15, hi = lane >> 4;
  const int n0 = (threadIdx.x >> 5) << 4;

  v8f acc[MSUB];
#pragma unroll
  for (int t = 0; t < MSUB; ++t) acc[t] = (v8f){};

  // first half of K: X (+AX), B rows 0..127
#pragma unroll 2
  for (int k = 0; k < HH; k += 4) {
    const int kk = k + 2 * hi;
    v2f b = {Bs[kk * PADW + n0 + lo], Bs[kk * PADW + PADW + n0 + lo]};
#pragma unroll
    for (int t = 0; t < MSUB; ++t) {
      const size_t row = (size_t)(m0 + 16 * t + lo);
      v2f a = *(const v2f*)(X + row * HH + kk);
      if (AX) {
        v2f a2 = *(const v2f*)(AX + row * HH + kk);
        a.x += a2.x; a.y += a2.y;
      }
      acc[t] = wmma4(a, b, acc[t]);
    }
  }
  // second half of K: S (+AS), B rows 128..255
#pragma unroll 2
  for (int k = 0; k < HH; k += 4) {
    const int kk = k + 2 * hi;
    v2f b = {Bs[(HH + kk) * PADW + n0 + lo],
             Bs[(HH + kk) * PADW + PADW + n0 + lo]};
#pragma unroll
    for (int t = 0; t < MSUB; ++t) {
      const size_t row = (size_t)(m0 + 16 * t + lo);
      v2f a = *(const v2f*)(S + row * HH + kk);
      if (AS) {
        v2f a2 = *(const v2f*)(AS + row * HH + kk);
        a.x += a2.x; a.y += a2.y;
      }
      acc[t] = wmma4(a, b, acc[t]);
    }
  }

  float bv = bias ? bias[n0 + lo] : 0.0f;
#pragma unroll
  for (int t = 0; t < MSUB; ++t)
#pragma unroll
    for (int j = 0; j < 8; ++j) {
      float v = acc[t][j] + bv;
      if (act) v = fmaxf(v, 0.0f);
      D[(size_t)(m0 + 16 * t + j + 8 * hi) * HH + n0 + lo] = v;
    }
}

// ---- D = A[N,16] @ B[16,128] + bias  (embedding of s; K=16, tiny) ----------
__global__ __launch_bounds__(256) void gemm_emb(
    const float* __restrict__ A, const float* __restrict__ B,
    const float* __restrict__ bias, float* __restrict__ D) {
  const int lane = threadIdx.x & 31;
  const int lo = lane & 15, hi = lane >> 4;
  const int n0 = (threadIdx.x >> 5) << 4;
  const int m0 = blockIdx.x << 4;

  const float* arow = A + (size_t)(m0 + lo) * 16 + 2 * hi;
  const float* bcol = B + (size_t)(2 * hi) * HH + n0 + lo;
  v8f c = {};
#pragma unroll
  for (int k = 0; k < 16; k += 4) {
    v2f a = *(const v2f*)(arow + k);
    v2f b = {bcol[(size_t)k * HH], bcol[(size_t)k * HH + HH]};
    c = wmma4(a, b, c);
  }
  float bv = bias[n0 + lo];
#pragma unroll
  for (int j = 0; j < 8; ++j)
    D[(size_t)(m0 + j + 8 * hi) * HH + n0 + lo] = c[j] + bv;
}

// ---- helpers ---------------------------------------------------------------
__global__ void zerok(float* __restrict__ p, long long n) {
  long long i = (long long)blockIdx.x * blockDim.x + threadIdx.x;
  long long st = (long long)gridDim.x * blockDim.x;
  for (; i < n; i += st) p[i] = 0.0f;
}

__global__ void deg_scatter(const int* __restrict__ ei, float* __restrict__ deg,
                            int E) {
  int e = blockIdx.x * blockDim.x + threadIdx.x;
  if (e < E) atomicAdd(&deg[ei[E + e]], 1.0f);   // dst = ei[1][e]
}

__global__ void dinv_kernel(float* __restrict__ d, int N) {
  int n = blockIdx.x * blockDim.x + threadIdx.x;
  if (n < N) d[n] = rsqrtf(d[n] + 1.0f);         // deg includes +1 self loop
}

// agg_x[dst] += x[src], agg_s[dst] += s[src] ; one thread per (edge, feature)
__global__ __launch_bounds__(256) void gin_scatter(
    const int* __restrict__ ei, const float* __restrict__ x,
    const float* __restrict__ s, float* __restrict__ ax,
    float* __restrict__ as_, int E) {
  long long tid = (long long)blockIdx.x * blockDim.x + threadIdx.x;
  if (tid >= (long long)E * HH) return;
  int e = (int)(tid >> 7);
  int f = (int)(tid & (HH - 1));
  int src = ei[e], dst = ei[E + e];
  atomicAdd(&ax[(size_t)dst * HH + f], x[(size_t)src * HH + f]);
  atomicAdd(&as_[(size_t)dst * HH + f], s[(size_t)src * HH + f]);
}

// sagg[dst] += hs[src] * dinv[src]*dinv[dst]
__global__ __launch_bounds__(256) void gcn_scatter(
    const int* __restrict__ ei, const float* __restrict__ hs,
    const float* __restrict__ dinv, float* __restrict__ sagg, int E) {
  long long tid = (long long)blockIdx.x * blockDim.x + threadIdx.x;
  if (tid >= (long long)E * HH) return;
  int e = (int)(tid >> 7);
  int f = (int)(tid & (HH - 1));
  int src = ei[e], dst = ei[E + e];
  float w = dinv[src] * dinv[dst];
  atomicAdd(&sagg[(size_t)dst * HH + f], hs[(size_t)src * HH + f] * w);
}

// s = tanh(sagg + hs*dinv^2 + b)
__global__ __launch_bounds__(256) void s_update(
    float* __restrict__ s, const float* __restrict__ sagg,
    const float* __restrict__ hs, const float* __restrict__ dinv,
    const float* __restrict__ b, int N) {
  long long tid = (long long)blockIdx.x * blockDim.x + threadIdx.x;
  if (tid >= (long long)N * HH) return;
  int n = (int)(tid >> 7);
  int f = (int)(tid & (HH - 1));
  float dv = dinv[n];
  s[tid] = tanhf(sagg[tid] + hs[tid] * dv * dv + b[f]);
}

// g[batch[n]] += y[n]
__global__ __launch_bounds__(256) void pool_kernel(
    const float* __restrict__ y, const int* __restrict__ batch,
    float* __restrict__ g, int N) {
  long long tid = (long long)blockIdx.x * blockDim.x + threadIdx.x;
  if (tid >= (long long)N * HH) return;
  int n = (int)(tid >> 7);
  int f = (int)(tid & (HH - 1));
  atomicAdd(&g[(size_t)batch[n] * HH + f], y[tid]);
}

// per-graph head: relu(g@post_w+post_b) @ ro_w + ro_b, then log_softmax (C=10)
__global__ __launch_bounds__(128) void head_kernel(
    const float* __restrict__ g, const float* __restrict__ pw,
    const float* __restrict__ pb, const float* __restrict__ rw,
    const float* __restrict__ rb, float* __restrict__ out, int C) {
  __shared__ float t[HH];
  __shared__ float lg[32];
  __shared__ float red[2];
  const int row = blockIdx.x;
  const int j = threadIdx.x;
  const float* gr = g + (size_t)row * HH;
  float acc = pb[j];
  for (int k = 0; k < HH; ++k) acc += gr[k] * pw[(size_t)k * HH + j];
  t[j] = fmaxf(acc, 0.0f);
  __syncthreads();
  if (j < C) {
    float a = rb[j];
    for (int k = 0; k < HH; ++k) a += t[k] * rw[(size_t)k * C + j];
    lg[j] = a;
  }
  __syncthreads();
  if (j == 0) {
    float mx = lg[0];
    for (int c = 1; c < C; ++c) mx = fmaxf(mx, lg[c]);
    float se = 0.0f;
    for (int c = 0; c < C; ++c) se += expf(lg[c] - mx);
    red[0] = mx;
    red[1] = logf(se);
  }
  __syncthreads();
  if (j < C) out[(size_t)row * C + j] = lg[j] - red[0] - red[1];
}

// ---------------------------------------------------------------------------
extern "C" void kernel_launch(void* const* d_in, const int* in_sizes, int n_in,
                              void* d_out, int out_size, void* d_ws,
                              size_t ws_size, hipStream_t stream) {
  const float* x_in   = (const float*)d_in[0];
  const float* s_in   = (const float*)d_in[1];
  const int*   ei     = (const int*)d_in[2];
  const int*   batch  = (const int*)d_in[3];
  const float* pre_w  = (const float*)d_in[4];
  const float* pre_b  = (const float*)d_in[5];
  const float* emb_w  = (const float*)d_in[6];
  const float* emb_b  = (const float*)d_in[7];
  const float* gin_w1 = (const float*)d_in[8];
  const float* gin_b1 = (const float*)d_in[9];
  const float* gin_w2 = (const float*)d_in[10];
  const float* gin_b2 = (const float*)d_in[11];
  const float* gcn_w  = (const float*)d_in[12];
  const float* gcn_b  = (const float*)d_in[13];
  const float* whp_w  = (const float*)d_in[14];
  const float* whp_b  = (const float*)d_in[15];
  const float* post_w = (const float*)d_in[16];
  const float* post_b = (const float*)d_in[17];
  const float* ro_w   = (const float*)d_in[18];
  const float* ro_b   = (const float*)d_in[19];
  float* out = (float*)d_out;

  const int N = in_sizes[0] / HH;       // 50000
  const int E = in_sizes[2] / 2;        // 800000
  const int G = 256;
  const int C = in_sizes[18] / HH;      // 10
  const int L = 3;

  // workspace layout (fp32)
  const size_t nf = (size_t)N * HH;
  float* xb   = (float*)d_ws;           // node features x   [N,H]
  float* sb   = xb + nf;                // node features s   [N,H]
  float* ax   = sb + nf;                // GIN agg of x / later hs
  float* as_  = ax + nf;                // GIN agg of s / later sagg
  float* t1   = as_ + nf;               // MLP hidden / whp output
  float* dinv = t1 + nf;                // deg -> 1/sqrt(deg+1)  [N]
  float* g    = dinv + N;               // pooled graphs [G,H]

  const int mt80 = N / (16 * MSUB);     // 625 row stripes of 80
  const int mt16 = N / 16;              // 3125 tiles of 16 (emb)
  const long long ef = (long long)E * HH;
  const int efb = (int)((ef + 255) / 256);
  const long long nfl = (long long)nf;
  const int nfb = (int)((nfl + 255) / 256);

  // degree / norm
  zerok<<<1024, 256, 0, stream>>>(dinv, N);
  deg_scatter<<<(E + 255) / 256, 256, 0, stream>>>(ei, dinv, E);
  dinv_kernel<<<(N + 255) / 256, 256, 0, stream>>>(dinv, N);

  // input projections
  gemm128<<<mt80, 256, 0, stream>>>(x_in, pre_w, pre_b, xb, 0);
  gemm_emb<<<mt16, 256, 0, stream>>>(s_in, emb_w, emb_b, sb);

  for (int i = 0; i < L; ++i) {
    // GIN aggregation of concat(x,s)
    zerok<<<4096, 256, 0, stream>>>(ax, 2 * nfl);   // ax and as_ contiguous
    gin_scatter<<<efb, 256, 0, stream>>>(ei, xb, sb, ax, as_, E);
    // GIN MLP layer 1: relu([x+ax | s+as] @ W1 + b1) -> t1
    gemm_cat<<<mt80, 256, 0, stream>>>(xb, ax, sb, as_,
                                       gin_w1 + (size_t)i * 2 * HH * HH,
                                       gin_b1 + (size_t)i * HH, t1, 1);
    // GCN: hs = s @ gcn_w  (reuse ax; gemm_cat above already consumed it)
    gemm128<<<mt80, 256, 0, stream>>>(sb, gcn_w + (size_t)i * HH * HH, nullptr,
                                      ax, 0);
    // GIN MLP layer 2: x = relu(t1 @ W2 + b2)
    gemm128<<<mt80, 256, 0, stream>>>(t1, gin_w2 + (size_t)i * HH * HH,
                                      gin_b2 + (size_t)i * HH, xb, 1);
    // GCN aggregation + update
    zerok<<<4096, 256, 0, stream>>>(as_, nfl);
    gcn_scatter<<<efb, 256, 0, stream>>>(ei, ax, dinv, as_, E);
    s_update<<<nfb, 256, 0, stream>>>(sb, as_, ax, dinv,
                                      gcn_b + (size_t)i * HH, N);
  }

  // whp: [x | s] @ whp_w + whp_b -> t1
  gemm_cat<<<mt80, 256, 0, stream>>>(xb, nullptr, sb, nullptr, whp_w, whp_b,
                                     t1, 0);
  // global add pool
  zerok<<<256, 256, 0, stream>>>(g, (long long)G * HH);
  pool_kernel<<<nfb, 256, 0, stream>>>(t1, batch, g, N);
  // post MLP + readout + log_softmax
  head_kernel<<<G, 128, 0, stream>>>(g, post_w, post_b, ro_w, ro_b, out, C);
}